// ETF_53738630807791
// MI455X (gfx1250) — compile-verified
//
#include <hip/hip_runtime.h>
#include <hip/hip_bf16.h>
#include <stdint.h>

#define H 1024
#define W 1024
#define NB 2
#define MU 5
#define TS 32              // output tile (32x32)
#define SH (TS + 2 * MU)   // 42 (tile + halo)
#define SP (SH + 1)        // 43, padded pitch -> conflict-free rows & cols
#define HW (H * W)

// ---------------------------------------------------------------------------
// CDNA5 async global->LDS copy (VGLOBAL GLOBAL_LOAD_ASYNC_TO_LDS_B32, op 96).
// Per-lane: copies one dword from vaddr into LDS at vdst; tracked by ASYNCcnt.
// Generic LDS pointers truncate to the wave-relative LDS byte offset
// (flat-LDS aperture maps addr[31:0] -> LDS offset), which is what VDST wants.
// ---------------------------------------------------------------------------
__device__ __forceinline__ void async_ld_f32(unsigned lds_addr, const float* g) {
    asm volatile("global_load_async_to_lds_b32 %0, %1, off"
                 :
                 : "v"(lds_addr), "v"((unsigned long long)(size_t)g)
                 : "memory");
}
__device__ __forceinline__ void wait_async0() {
    asm volatile("s_wait_asynccnt 0" ::: "memory");
}

// Hardware approximations (v_rcp_f32 / v_rsq_f32, ~1 ulp) instead of IEEE
// div/sqrt lowering — the ETF weights are re-normalized every pass, so the
// approximation error is irrelevant, and this removes ~8 VALU ops per tap.
__device__ __forceinline__ float fast_rcp(float x) { return __builtin_amdgcn_rcpf(x); }
__device__ __forceinline__ float fast_rsq(float x) { return __builtin_amdgcn_rsqf(x); }

// ---------------------------------------------------------------------------
// Kernel 1: Sobel -> mag, tang; block-max of mag -> partials
// ---------------------------------------------------------------------------
__global__ __launch_bounds__(256)
void etf_init(const float* __restrict__ in, float* __restrict__ tang,
              float* __restrict__ mag, float* __restrict__ partial) {
    __shared__ float red[256];
    const int tid = threadIdx.x;
    const int idx = blockIdx.x * 256 + tid;          // 0 .. 2*HW-1
    const int b = idx >> 20;                          // HW = 1<<20
    const int p = idx & (HW - 1);
    const int y = p >> 10, x = p & (W - 1);
    const float* im = in + (size_t)b * HW;

    auto ld = [&](int yy, int xx) -> float {
        return (yy < 0 || yy >= H || xx < 0 || xx >= W) ? 0.f : im[yy * W + xx];
    };
    float a = ld(y - 1, x - 1), b1 = ld(y - 1, x), c = ld(y - 1, x + 1);
    float d = ld(y,     x - 1),                    e = ld(y,     x + 1);
    float f = ld(y + 1, x - 1), g = ld(y + 1, x),  h = ld(y + 1, x + 1);

    float g0 = (f + 2.f * g + h) - (a + 2.f * b1 + c);   // conv with k
    float g1 = (c + 2.f * e + h) - (a + 2.f * d + f);    // conv with k.T
    float m = sqrtf(g0 * g0 + g1 * g1);
    float inv = m > 0.f ? fast_rcp(m) : 0.f;              // g0=g1=0 when m==0

    mag[idx] = m;
    tang[((size_t)b * 2    ) * HW + p] = -g1 * inv;       // tx
    tang[((size_t)b * 2 + 1) * HW + p] =  g0 * inv;       // ty

    red[tid] = m;
    __syncthreads();
    for (int s = 128; s > 0; s >>= 1) {
        if (tid < s) red[tid] = fmaxf(red[tid], red[tid + s]);
        __syncthreads();
    }
    if (tid == 0) partial[blockIdx.x] = red[0];
}

// ---------------------------------------------------------------------------
// Kernel 2: reduce partial maxima -> invM = 1/max
// ---------------------------------------------------------------------------
__global__ __launch_bounds__(256)
void etf_reduce(const float* __restrict__ partial, float* __restrict__ invM, int n) {
    __shared__ float red[256];
    float m = 0.f;
    for (int i = threadIdx.x; i < n; i += 256) m = fmaxf(m, partial[i]);
    red[threadIdx.x] = m;
    __syncthreads();
    for (int s = 128; s > 0; s >>= 1) {
        if (threadIdx.x < s)
            red[threadIdx.x] = fmaxf(red[threadIdx.x], red[threadIdx.x + s]);
        __syncthreads();
    }
    if (threadIdx.x == 0) *invM = red[0] > 0.f ? 1.f / red[0] : 1.f;
}

// ---------------------------------------------------------------------------
// Kernel 3: one ETF smoothing pass (DX,DY select orientation).
// 32x32 output tile per block, 42x42 halo tile staged in LDS via async copies.
// OOB halo cells are zero-filled (zero tangent => zero weight, matches pad).
// ---------------------------------------------------------------------------
template <int DX, int DY>
__global__ __launch_bounds__(256)
void etf_smooth(const float* __restrict__ tin, const float* __restrict__ mag,
                const float* __restrict__ invMp, float* __restrict__ tout) {
    __shared__ float s_tx[SH][SP];
    __shared__ float s_ty[SH][SP];
    __shared__ float s_mg[SH][SP];

    const int b  = blockIdx.z;
    const int x0 = blockIdx.x * TS;
    const int y0 = blockIdx.y * TS;
    const int tid = threadIdx.y * 32 + threadIdx.x;
    const float invM = *invMp;

    const float* tx_p = tin + ((size_t)b * 2    ) * HW;
    const float* ty_p = tin + ((size_t)b * 2 + 1) * HW;
    const float* mg_p = mag + (size_t)b * HW;

    // Stage halo tile: async global->LDS for in-bounds, ds_store 0 for OOB.
    for (int i = tid; i < SH * SH; i += 256) {
        const int lx = i % SH, ly = i / SH;
        const int gx = x0 - MU + lx, gy = y0 - MU + ly;
        if (gx >= 0 && gx < W && gy >= 0 && gy < H) {
            const size_t off = (size_t)gy * W + gx;
            async_ld_f32((unsigned)(size_t)&s_tx[ly][lx], tx_p + off);
            async_ld_f32((unsigned)(size_t)&s_ty[ly][lx], ty_p + off);
            async_ld_f32((unsigned)(size_t)&s_mg[ly][lx], mg_p + off);
        } else {
            s_tx[ly][lx] = 0.f;
            s_ty[ly][lx] = 0.f;
            s_mg[ly][lx] = 0.f;
        }
    }
    wait_async0();
    __syncthreads();

    const int lx = threadIdx.x + MU;
#pragma unroll
    for (int r = 0; r < 4; ++r) {
        const int oy = threadIdx.y + r * 8;
        const int ly = oy + MU;
        const float cx = s_tx[ly][lx];
        const float cy = s_ty[ly][lx];
        const float cm = s_mg[ly][lx];
        float ax = 0.f, ay = 0.f;
#pragma unroll
        for (int k = -MU; k <= MU; ++k) {
            const float sx = s_tx[ly + k * DY][lx + k * DX];
            const float sy = s_ty[ly + k * DY][lx + k * DX];
            const float sm = s_mg[ly + k * DY][lx + k * DX];
            const float dot = cx * sx + cy * sy;
            // (tanh(d)+1)/2 == e^{2d}/(e^{2d}+1); d in [-1,1] so no overflow.
            const float e = __expf(2.f * (sm - cm) * invM);
            const float w = dot * e * fast_rcp(e + 1.f);   // v_exp + v_rcp + muls
            ax += sx * w;
            ay += sy * w;
        }
        const float s = ax * ax + ay * ay;
        const float inv = s > 0.f ? fast_rsq(s) : 0.f;     // s==0 -> (0,0), matches ref
        const size_t o = (size_t)(y0 + oy) * W + (x0 + threadIdx.x);
        tout[((size_t)b * 2    ) * HW + o] = ax * inv;
        tout[((size_t)b * 2 + 1) * HW + o] = ay * inv;
    }
}

// ---------------------------------------------------------------------------
// Workspace layout (floats): [0,2M) mag | [2M,6M) tang ping | [6M,+8192) partials
//                            | +1 invM     (~24 MB total)
// ---------------------------------------------------------------------------
extern "C" void kernel_launch(void* const* d_in, const int* in_sizes, int n_in,
                              void* d_out, int out_size, void* d_ws, size_t ws_size,
                              hipStream_t stream) {
    (void)in_sizes; (void)n_in; (void)out_size; (void)ws_size;
    const float* in = (const float*)d_in[0];
    float* out = (float*)d_out;                 // [2][2][H][W]
    float* ws  = (float*)d_ws;

    float* mag     = ws;                         // 2*HW
    float* tping   = ws + (size_t)NB * HW;       // 4*HW (2*2*HW floats)
    float* partial = ws + (size_t)6 * HW;        // 8192
    float* invM    = partial + 8192;             // 1

    etf_init<<<(NB * HW) / 256, 256, 0, stream>>>(in, out, mag, partial);
    etf_reduce<<<1, 256, 0, stream>>>(partial, invM, (NB * HW) / 256);

    dim3 grid(W / TS, H / TS, NB);
    dim3 blk(32, 8);
    for (int p = 0; p < 6; ++p) {
        const float* src = (p & 1) ? tping : out;
        float*       dst = (p & 1) ? out : tping;
        if (p & 1)
            etf_smooth<1, 0><<<grid, blk, 0, stream>>>(src, mag, invM, dst); // horizontal
        else
            etf_smooth<0, 1><<<grid, blk, 0, stream>>>(src, mag, invM, dst); // vertical
    }
}